// MyGCN_2027224564199
// MI455X (gfx1250) — compile-verified
//
#include <hip/hip_runtime.h>

#define NN 100000
#define NE 1600000
#define F1 64

typedef __attribute__((ext_vector_type(2))) float v2f;
typedef __attribute__((ext_vector_type(8))) float v8f;

__device__ __forceinline__ void atomAddF(float* p, float v) {
    (void)__hip_atomic_fetch_add(p, v, __ATOMIC_RELAXED, __HIP_MEMORY_SCOPE_AGENT);
}

__global__ void k_zero(float* __restrict__ p, int n) {
    int i = blockIdx.x * blockDim.x + threadIdx.x;
    if (i < n) p[i] = 0.0f;
}

// deg[col[e]] += 1  (f32 atomics; deg values are exact small integers in f32)
__global__ void k_degree(const int* __restrict__ col, float* __restrict__ deg) {
    int e = blockIdx.x * blockDim.x + threadIdx.x;
    if (e < NE) atomAddF(&deg[col[e]], 1.0f);
}

// dis[i] = deg>0 ? deg^-0.5 : 0   (exact sqrt+div, no hw-approx rsq)
__global__ void k_inv_sqrt(float* __restrict__ dis) {
    int i = blockIdx.x * blockDim.x + threadIdx.x;
    if (i < NN) {
        float d = dis[i];
        dis[i] = (d > 0.0f) ? (1.0f / sqrtf(d)) : 0.0f;
    }
}

// h[N,64] = x[N,2] @ W1[2,64] via V_WMMA_F32_16X16X4_F32 (K=2 padded to 4).
// One wave handles a 16-node tile; 4 WMMA issues cover the 64 output columns.
// A layout (16x4 f32): lanes 0-15 hold K=0,1 (vgpr0,1); lanes 16-31 hold K=2,3 (zero pad).
// B layout (4x16 f32): lanes 0-15 hold rows K=0,1; lanes 16-31 rows K=2,3 (zero pad).
__global__ void k_gemm1_wmma(const float* __restrict__ x,
                             const float* __restrict__ W1,
                             float* __restrict__ h) {
    const int lane   = threadIdx.x & 31;
    const int waveIb = threadIdx.x >> 5;
    const int tile   = blockIdx.x * (blockDim.x >> 5) + waveIb;
    const int nTiles = NN / 16;            // 6250, exact
    if (tile >= nTiles) return;            // wave-uniform: EXEC all-1s inside
    const int mBase = tile * 16;

    v2f a = {0.0f, 0.0f};
    if (lane < 16) {
        const float* xp = x + (mBase + lane) * 2;
        a.x = xp[0];                        // K=0
        a.y = xp[1];                        // K=1
    }
    for (int nb = 0; nb < 4; ++nb) {
        const int n0 = nb * 16;
        v2f b = {0.0f, 0.0f};
        if (lane < 16) {
            b.x = W1[0 * F1 + n0 + lane];   // K=0 row
            b.y = W1[1 * F1 + n0 + lane];   // K=1 row
        }
        v8f c = {};
        c = __builtin_amdgcn_wmma_f32_16x16x4_f32(
                /*neg_a=*/false, a, /*neg_b=*/false, b,
                /*c_mod=*/(short)0, c, /*reuse_a=*/false, /*reuse_b=*/false);
        // C/D layout: vgpr r -> M = r + (lane>=16 ? 8 : 0), N = n0 + (lane & 15)
        const int nCol = n0 + (lane & 15);
        const int rOff = (lane >= 16) ? 8 : 0;
#pragma unroll
        for (int r = 0; r < 8; ++r) {
            h[(mBase + r + rOff) * F1 + nCol] = c[r];
        }
    }
}

// Edge pass 1: 16 threads per edge, float4 gather of h[row], scale, 4 f32 atomics.
__global__ void k_edge1(const int* __restrict__ row, const int* __restrict__ col,
                        const float* __restrict__ dis, const float* __restrict__ h,
                        float* __restrict__ agg) {
    long long gid = (long long)blockIdx.x * blockDim.x + threadIdx.x;
    int e = (int)(gid >> 4);
    if (e >= NE) return;
    int f = (int)(gid & 15) << 2;
    int r = row[e], c = col[e];
    float w = dis[r] * dis[c];
    const float4 hv = *reinterpret_cast<const float4*>(h + (long long)r * F1 + f);
    float* ap = agg + (long long)c * F1 + f;
    atomAddF(ap + 0, hv.x * w);
    atomAddF(ap + 1, hv.y * w);
    atomAddF(ap + 2, hv.z * w);
    atomAddF(ap + 3, hv.w * w);
}

// Fused: h2[n] = relu(agg1[n,:] + b1) . W2[:,0]   (h1 never materialized)
__global__ void k_node2(const float* __restrict__ agg1, const float* __restrict__ b1,
                        const float* __restrict__ W2, float* __restrict__ h2) {
    int n = blockIdx.x * blockDim.x + threadIdx.x;
    if (n >= NN) return;
    const float* ap = agg1 + (long long)n * F1;
    float acc = 0.0f;
#pragma unroll
    for (int f = 0; f < F1; ++f) {
        float v = ap[f] + b1[f];
        v = v > 0.0f ? v : 0.0f;
        acc += v * W2[f];
    }
    h2[n] = acc;
}

// Edge pass 2: out[col] += h2[row] * norm
__global__ void k_edge2(const int* __restrict__ row, const int* __restrict__ col,
                        const float* __restrict__ dis, const float* __restrict__ h2,
                        float* __restrict__ out) {
    int e = blockIdx.x * blockDim.x + threadIdx.x;
    if (e < NE) {
        int r = row[e], c = col[e];
        atomAddF(&out[c], h2[r] * dis[r] * dis[c]);
    }
}

__global__ void k_final(float* __restrict__ out, const float* __restrict__ b2) {
    int i = blockIdx.x * blockDim.x + threadIdx.x;
    if (i < NN) {
        float v = out[i] + b2[0];
        out[i] = v > 0.0f ? v : 0.0f;
    }
}

extern "C" void kernel_launch(void* const* d_in, const int* in_sizes, int n_in,
                              void* d_out, int out_size, void* d_ws, size_t ws_size,
                              hipStream_t stream) {
    const float* x  = (const float*)d_in[0];
    const int*   ei = (const int*)  d_in[1];   // [2, NE] int32 per harness contract
    const float* W1 = (const float*)d_in[2];
    const float* b1 = (const float*)d_in[3];
    const float* W2 = (const float*)d_in[4];
    const float* b2 = (const float*)d_in[5];
    float* out = (float*)d_out;
    float* ws  = (float*)d_ws;

    const int* row = ei;            // source
    const int* col = ei + NE;       // target

    // Workspace layout (floats): dis[N] | h[N*64] | agg1[N*64]; h2 aliases h
    float* dis  = ws;
    float* h    = ws + NN;
    float* agg1 = h + (size_t)NN * F1;
    float* h2   = h;                // h is dead after k_edge1; reuse for h2

    const int T = 256;

    // Zero accumulation targets (d_out/d_ws are poisoned by the harness)
    k_zero<<<(NN + T - 1) / T, T, 0, stream>>>(dis, NN);
    k_zero<<<(NN * F1 + T - 1) / T, T, 0, stream>>>(agg1, NN * F1);
    k_zero<<<(NN + T - 1) / T, T, 0, stream>>>(out, NN);

    // Degrees + symmetric normalization
    k_degree<<<(NE + T - 1) / T, T, 0, stream>>>(col, dis);
    k_inv_sqrt<<<(NN + T - 1) / T, T, 0, stream>>>(dis);

    // Layer 1 GEMM via f32 WMMA: 6250 tiles, 8 waves per 256-thread block
    k_gemm1_wmma<<<(6250 + 7) / 8, T, 0, stream>>>(x, W1, h);

    // Layer 1 aggregation: 16 threads/edge
    long long e1Threads = (long long)NE * 16;                 // 25.6M
    k_edge1<<<(int)((e1Threads + T - 1) / T), T, 0, stream>>>(row, col, dis, h, agg1);

    // relu + bias + dot(W2) fused per node
    k_node2<<<(NN + T - 1) / T, T, 0, stream>>>(agg1, b1, W2, h2);

    // Layer 2 aggregation: 1 thread/edge
    k_edge2<<<(NE + T - 1) / T, T, 0, stream>>>(row, col, dis, h2, out);

    // Final bias + relu
    k_final<<<(NN + T - 1) / T, T, 0, stream>>>(out, b2);
}